// GCNFlowPredictor_61804579390070
// MI455X (gfx1250) — compile-verified
//
#include <hip/hip_runtime.h>
#include <hip/hip_bf16.h>
#include <math.h>

#define N_NODES 100000
#define E_EDGES 3200000
#define F_IN    64
#define HFEAT   48
#define N_OUT   64
#define EB      32          // edges staged per block in the scatter kernel

typedef __attribute__((ext_vector_type(2))) float v2f;
typedef __attribute__((ext_vector_type(8))) float v8f;

// ---------------- degree / normalization ----------------

__global__ void zero_f32_kernel(float* __restrict__ p, int n) {
    int i = blockIdx.x * blockDim.x + threadIdx.x;
    if (i < n) p[i] = 0.0f;
}

__global__ void count_deg_kernel(const long long* __restrict__ col,
                                 float* __restrict__ cnt, int e) {
    int i = blockIdx.x * blockDim.x + threadIdx.x;
    if (i < e) atomicAdd(&cnt[(int)col[i]], 1.0f);
}

__global__ void finish_dis_kernel(float* __restrict__ dis, int n) {
    int i = blockIdx.x * blockDim.x + threadIdx.x;
    if (i < n) dis[i] = rsqrtf(dis[i] + 1.0f);   // +1 self-loop; always > 0
}

// ---------------- dense transform: T = relu?(X) @ W via V_WMMA_F32_16X16X4_F32
// X: [N, K] row-major, W: [K, 48] row-major, T: [N, 48] row-major.
// One wave computes a 16-row x 48-col stripe. K stepped by 4.
// RELU applies ReLU to the A operand on load (fuses the activation of the
// previous layer into this GEMM's input path).

template<int K, bool RELU>
__global__ void gemm_wmma_kernel(const float* __restrict__ X,
                                 const float* __restrict__ W,
                                 float* __restrict__ T,
                                 int nRowBlocks) {
    int lane = threadIdx.x & 31;
    int wave = blockIdx.x * (blockDim.x >> 5) + (threadIdx.x >> 5);
    if (wave >= nRowBlocks) return;              // uniform per wave -> EXEC all-1s below

    const int half = lane >> 4;                  // 0: lanes 0-15, 1: lanes 16-31
    const int m    = lane & 15;                  // A row / B,C,D column within tile
    const int kb   = half * 2;                   // K sub-offset per lane half

    v8f c0 = {}; v8f c1 = {}; v8f c2 = {};
    const float* xrow = X + (wave * 16 + m) * K;

    for (int k = 0; k < K; k += 4) {
        // A 16x4 fp32: VGPR0 -> K = kb, VGPR1 -> K = kb+1
        float ax = xrow[k + kb];
        float ay = xrow[k + kb + 1];
        if (RELU) { ax = fmaxf(ax, 0.0f); ay = fmaxf(ay, 0.0f); }
        v2f a; a.x = ax; a.y = ay;
        // B 4x16 fp32: same K-per-lane-half striping, N = m
        const float* w0 = W + (k + kb) * HFEAT;
        const float* w1 = W + (k + kb + 1) * HFEAT;
        v2f b0, b1, b2;
        b0.x = w0[m];       b0.y = w1[m];
        b1.x = w0[16 + m];  b1.y = w1[16 + m];
        b2.x = w0[32 + m];  b2.y = w1[32 + m];

        c0 = __builtin_amdgcn_wmma_f32_16x16x4_f32(false, a, false, b0, (short)0, c0, false, false);
        c1 = __builtin_amdgcn_wmma_f32_16x16x4_f32(false, a, false, b1, (short)0, c1, false, false);
        c2 = __builtin_amdgcn_wmma_f32_16x16x4_f32(false, a, false, b2, (short)0, c2, false, false);
    }

    // D 16x16 fp32: element r -> M = r + 8*half, N = m
    const int mbase = wave * 16 + half * 8;
    #pragma unroll
    for (int r = 0; r < 8; ++r) {
        float* trow = T + (mbase + r) * HFEAT;
        trow[m]      = c0[r];
        trow[16 + m] = c1[r];
        trow[32 + m] = c2[r];
    }
}

// ---------------- aggregation ----------------

// out[n,f] = t[n,f]*dis[n]^2 + bias[f]   (self-loop term + bias, also zero-inits)
__global__ void init_self_kernel(const float* __restrict__ t,
                                 const float* __restrict__ dis,
                                 const float* __restrict__ bias,
                                 float* __restrict__ out, int total) {
    int i = blockIdx.x * blockDim.x + threadIdx.x;
    if (i < total) {
        int node = i / HFEAT;
        int f    = i - node * HFEAT;
        float d  = dis[node];
        out[i] = t[i] * d * d + bias[f];
    }
}

// out[col,f] += t[row,f] * dis[row] * dis[col]
// Block stages EB=32 edges (indices pre-scaled + norm precomputed) in LDS,
// then 256 threads sweep the 32*48 = 1536 (edge, feature) pairs in 6 passes:
// exactly one coalesced load + one f32 atomic per pair.
// Requires: gridDim.x * EB == E (3.2M % 32 == 0, no tail).
__global__ void scatter_edges_kernel(const long long* __restrict__ row,
                                     const long long* __restrict__ col,
                                     const float* __restrict__ t,
                                     const float* __restrict__ dis,
                                     float* __restrict__ out) {
    __shared__ int   sRow[EB];
    __shared__ int   sCol[EB];
    __shared__ float sNorm[EB];

    const int tid = threadIdx.x;
    const long long ebase = (long long)blockIdx.x * EB;
    if (tid < EB) {
        int r = (int)row[ebase + tid];
        int c = (int)col[ebase + tid];
        sRow[tid]  = r * HFEAT;
        sCol[tid]  = c * HFEAT;
        sNorm[tid] = dis[r] * dis[c];
    }
    __syncthreads();

    #pragma unroll
    for (int i = 0; i < (EB * HFEAT) / 256; ++i) {      // 6 passes
        int idx = i * 256 + tid;                         // 0 .. 1535
        int le  = idx / HFEAT;
        int f   = idx - le * HFEAT;
        atomicAdd(&out[sCol[le] + f], t[sRow[le] + f] * sNorm[le]);
    }
}

// ---------------- head: sigmoid(relu(h)[outfall] @ Wl + bl) ----------------

__global__ void head_kernel(const float* __restrict__ h,
                            const long long* __restrict__ outfall,
                            const float* __restrict__ Wl,
                            const float* __restrict__ bl,
                            float* __restrict__ out, int nout) {
    int i = blockIdx.x * blockDim.x + threadIdx.x;
    if (i >= nout) return;
    int node = (int)outfall[i];
    const float* hr = h + node * HFEAT;
    float acc = bl[0];
    #pragma unroll
    for (int j = 0; j < HFEAT; ++j) acc += fmaxf(hr[j], 0.0f) * Wl[j];
    out[i] = 1.0f / (1.0f + expf(-acc));
}

// ---------------- launch ----------------

extern "C" void kernel_launch(void* const* d_in, const int* in_sizes, int n_in,
                              void* d_out, int out_size, void* d_ws, size_t ws_size,
                              hipStream_t stream) {
    const float*     x    = (const float*)d_in[0];
    const long long* eidx = (const long long*)d_in[1];   // [2, E]: row then col
    const long long* outf = (const long long*)d_in[2];
    const float*     W1   = (const float*)d_in[3];
    const float*     b1   = (const float*)d_in[4];
    const float*     W2   = (const float*)d_in[5];
    const float*     b2   = (const float*)d_in[6];
    const float*     Wl   = (const float*)d_in[7];
    const float*     bl   = (const float*)d_in[8];
    float* out = (float*)d_out;

    const long long* row = eidx;
    const long long* col = eidx + E_EDGES;

    // Workspace layout: dis [N] | bufA [N*48] | bufB [N*48]  (~39 MB)
    char* ws = (char*)d_ws;
    float* dis  = (float*)ws;
    size_t off  = ((size_t)N_NODES * 4 + 4095) & ~(size_t)4095;
    float* bufA = (float*)(ws + off);                  // GEMM output T
    off += ((size_t)N_NODES * HFEAT * 4 + 4095) & ~(size_t)4095;
    float* bufB = (float*)(ws + off);                  // aggregation output / h

    const int NF      = N_NODES * HFEAT;               // 4.8M
    const int nRowBlk = N_NODES / 16;                  // 6250 (exact)
    const int B       = 256;
    const int wavesPerBlk = B / 32;

    // 1) normalization coefficients: dis = rsqrt(in_deg(col) + 1)
    zero_f32_kernel<<<(N_NODES + B - 1) / B, B, 0, stream>>>(dis, N_NODES);
    count_deg_kernel<<<(E_EDGES + B - 1) / B, B, 0, stream>>>(col, dis, E_EDGES);
    finish_dis_kernel<<<(N_NODES + B - 1) / B, B, 0, stream>>>(dis, N_NODES);

    const int gemmGrid    = (nRowBlk + wavesPerBlk - 1) / wavesPerBlk;
    const int scatterGrid = E_EDGES / EB;              // 100000 (exact)

    // 2) layer 1: T = x @ W1 ; agg -> bufB (relu deferred to consumers)
    gemm_wmma_kernel<F_IN, false><<<gemmGrid, B, 0, stream>>>(x, W1, bufA, nRowBlk);
    init_self_kernel<<<(NF + B - 1) / B, B, 0, stream>>>(bufA, dis, b1, bufB, NF);
    scatter_edges_kernel<<<scatterGrid, B, 0, stream>>>(row, col, bufA, dis, bufB);

    // 3) layer 2: T = relu(h1) @ W2 (reads bufB, writes bufA), agg back into bufB
    gemm_wmma_kernel<HFEAT, true><<<gemmGrid, B, 0, stream>>>(bufB, W2, bufA, nRowBlk);
    init_self_kernel<<<(NF + B - 1) / B, B, 0, stream>>>(bufA, dis, b2, bufB, NF);
    scatter_edges_kernel<<<scatterGrid, B, 0, stream>>>(row, col, bufA, dis, bufB);

    // 4) head: sigmoid(relu(h2)[outfall] @ Wl + bl)
    head_kernel<<<1, N_OUT, 0, stream>>>(bufB, outf, Wl, bl, out, N_OUT);
}